// DCT2DSpatialTransformLayer_52055003627865
// MI455X (gfx1250) — compile-verified
//
#include <hip/hip_runtime.h>
#include <math.h>

typedef __attribute__((ext_vector_type(2))) float v2f;
typedef __attribute__((ext_vector_type(8))) float v8f;

#define HH 128
#define WW 128
#define STX 140   // LDS row stride (dwords) for X/T buffer: 140%64=12 -> conflict-free patterns
#define STW 132   // LDS row stride (dwords) for W buffer:  132%64=4  -> conflict-free patterns

// ---------------------------------------------------------------------------
// Kernel 1: build the orthonormal DCT-II filter matrix W[v][j] into d_ws.
// W[v][j] = cos(pi*(0.5+j)*v/128)/sqrt(128) * (v!=0 ? sqrt(2) : 1)
// ---------------------------------------------------------------------------
__global__ void dct_filter_init(float* __restrict__ w) {
    int idx = blockIdx.x * blockDim.x + threadIdx.x;   // 0..16383
    if (idx >= HH * WW) return;
    int v = idx >> 7;
    int j = idx & 127;
    const float PI = 3.14159265358979323846f;
    float val = cosf(PI * (0.5f + (float)j) * (float)v * (1.0f / 128.0f))
              * 0.08838834764831845f;                  // 1/sqrt(128)
    if (v != 0) val *= 1.41421356237309515f;           // sqrt(2)
    w[idx] = val;
}

// ---------------------------------------------------------------------------
// Kernel 2: per-image 2D DCT. out = W * X * W^T using v_wmma_f32_16x16x4_f32.
// One block = one (b,c) image. 256 threads = 8 waves; wave w owns rows
// [16w, 16w+16) of the intermediate T and of the output.
// ---------------------------------------------------------------------------
__global__ __launch_bounds__(256)
void dct2d_main(const float* __restrict__ x,
                const float* __restrict__ wmat,
                float* __restrict__ out) {
    __shared__ float Xl[HH * STX];   // X, later reused as T (per-wave-strip rewrite)
    __shared__ float Wl[HH * STW];   // DCT matrix

    const int img = blockIdx.x;
    const float* xin = x + (size_t)img * (HH * WW);
    float* o = out + (size_t)img * (HH * WW);
    const int t = threadIdx.x;

    // Cooperative coalesced load: 4096 float4s of X and of W, 16 per thread.
    for (int i = 0; i < 16; ++i) {
        int e   = t + i * 256;        // float4 index 0..4095
        int row = e >> 5;             // 32 float4 per 128-wide row
        int c4  = (e & 31) << 2;      // starting float column
        float4 vx = ((const float4*)xin)[e];
        *(float4*)&Xl[row * STX + c4] = vx;
        float4 vw = ((const float4*)wmat)[e];
        *(float4*)&Wl[row * STW + c4] = vw;
    }
    __syncthreads();

    const int wave = t >> 5;
    const int lane = t & 31;
    const int m    = lane & 15;        // M (for A) / N (for B,C,D) within tile
    const int hi   = lane >> 4;        // lane-group selector
    const int kb   = hi << 1;          // K sub-offset: 0 or 2

    // ---- Load stage-1 A strip: X rows [16*wave, 16*wave+16) into registers.
    // A 16x4 f32 layout: lane&15 = M; reg r holds K = kb + r.
    v2f a[32];
    #pragma unroll
    for (int k = 0; k < 32; ++k)
        a[k] = *(const v2f*)&Xl[(16 * wave + m) * STX + 4 * k + kb];

    // ---- Stage 1: T = X * W^T.  B[k',n] = W[n_global, k'] -> rows of W.
    for (int n = 0; n < 8; ++n) {
        v8f acc = {};
        #pragma unroll
        for (int k = 0; k < 32; ++k) {
            v2f b = *(const v2f*)&Wl[(16 * n + m) * STW + 4 * k + kb];
            acc = __builtin_amdgcn_wmma_f32_16x16x4_f32(
                false, a[k], false, b, (short)0, acc, false, false);
        }
        // C/D layout: VGPR i -> row M = i + 8*hi, col N = lane&15.
        #pragma unroll
        for (int i = 0; i < 8; ++i)
            Xl[(16 * wave + i + 8 * hi) * STX + 16 * n + m] = acc[i];
    }
    __syncthreads();

    // ---- Load stage-2 A strip: W rows [16*wave, 16*wave+16).
    #pragma unroll
    for (int k = 0; k < 32; ++k)
        a[k] = *(const v2f*)&Wl[(16 * wave + m) * STW + 4 * k + kb];

    // ---- Stage 2: out = W * T.  B[k',n] = T[k', n_global] (T lives in Xl).
    for (int n = 0; n < 8; ++n) {
        v8f acc = {};
        #pragma unroll
        for (int k = 0; k < 32; ++k) {
            v2f b;
            b.x = Xl[(4 * k + kb)     * STX + 16 * n + m];
            b.y = Xl[(4 * k + kb + 1) * STX + 16 * n + m];
            acc = __builtin_amdgcn_wmma_f32_16x16x4_f32(
                false, a[k], false, b, (short)0, acc, false, false);
        }
        #pragma unroll
        for (int i = 0; i < 8; ++i)
            o[(16 * wave + i + 8 * hi) * WW + 16 * n + m] = acc[i];
    }
}

// ---------------------------------------------------------------------------
extern "C" void kernel_launch(void* const* d_in, const int* in_sizes, int n_in,
                              void* d_out, int out_size, void* d_ws, size_t ws_size,
                              hipStream_t stream) {
    const float* x = (const float*)d_in[0];
    float* out = (float*)d_out;
    float* wmat = (float*)d_ws;                  // 128*128*4 = 64KB scratch

    int total = in_sizes[0];                     // B*C*H*W
    int nimg  = total / (HH * WW);               // 4096

    dct_filter_init<<<(HH * WW + 255) / 256, 256, 0, stream>>>(wmat);
    dct2d_main<<<nimg, 256, 0, stream>>>(x, wmat, out);
}